// Cube_Norm_53876069761105
// MI455X (gfx1250) — compile-verified
//
#include <hip/hip_runtime.h>

// Problem constants (from the reference)
#define D            300        // EMBED_DIM
#define VEC_PER_ROW  (D / 4)    // 75 float4 chunks per row (1200 B, 16B-aligned)
#define CHUNK        256        // rows per LDS tile (== NODES_PER_GRAPH)
#define TPB          320        // 10 wave32s per workgroup
#define SCAN_MAX     1024
#define EPS          1e-12f
#define FLT_BIG      3.402823466e+38f

// ---------------------------------------------------------------------------
// CDNA5 async global -> LDS copy (16 bytes per lane-issue), GVS addressing:
//   mem_addr = SGPR_U64 + VGPR_I32 ; LDS dest = wave LDS base + VDST VGPR
// Tracked by ASYNCcnt (CDNA5 ISA).
// ---------------------------------------------------------------------------
__device__ __forceinline__ void async_copy_b128(unsigned lds_byte_off,
                                                unsigned g_byte_off,
                                                unsigned long long g_base) {
  asm volatile("global_load_async_to_lds_b128 %0, %1, %2"
               :
               : "v"(lds_byte_off), "v"(g_byte_off), "s"(g_base)
               : "memory");
}

__device__ __forceinline__ void wait_async0() {
#if __has_builtin(__builtin_amdgcn_s_wait_asynccnt)
  __builtin_amdgcn_s_wait_asynccnt(0);
#else
  asm volatile("s_wait_asynccnt 0" ::: "memory");
#endif
}

// ---------------------------------------------------------------------------
// Exclusive prefix sum of per-graph node counts (B <= 1024, single block).
// ---------------------------------------------------------------------------
__global__ __launch_bounds__(SCAN_MAX) void exclusive_scan_kernel(
    const int* __restrict__ counts, int* __restrict__ offsets, int B) {
  __shared__ int buf[SCAN_MAX];
  const int t = threadIdx.x;
  const int v = (t < B) ? counts[t] : 0;
  buf[t] = v;
  __syncthreads();
#pragma unroll
  for (int off = 1; off < SCAN_MAX; off <<= 1) {
    const int add = (t >= off) ? buf[t - off] : 0;
    __syncthreads();
    buf[t] += add;
    __syncthreads();
  }
  if (t < B) offsets[t] = buf[t] - v;  // exclusive prefix
}

// ---------------------------------------------------------------------------
// One workgroup per graph. Tile (<=300 KB) lives in LDS (320 KB/WGP):
//   A) async-DMA the graph tile global->LDS
//   B) per-dim min/max reduction from LDS, mid & 1/ldv into LDS
//   C) vectorized normalize LDS -> global (b128 stores)
// ---------------------------------------------------------------------------
__global__ __launch_bounds__(TPB) void seg_minmax_norm_kernel(
    const float* __restrict__ x, const int* __restrict__ counts,
    const int* __restrict__ offsets, float* __restrict__ out, int B) {
  __shared__ __align__(16) float tile[CHUNK * D];   // 307200 bytes
  __shared__ __align__(16) float s_mid[D + 4];
  __shared__ __align__(16) float s_inv[D + 4];

  const int b = blockIdx.x;
  if (b >= B) return;
  const int r0  = offsets[b];
  const int cnt = counts[b];
  if (cnt <= 0) return;

  const int tid = threadIdx.x;
  const unsigned long long g_in =
      (unsigned long long)x + (unsigned long long)r0 * (unsigned long long)(D * 4);
  float* const g_out = out + (long long)r0 * D;
  const unsigned lds0 = (unsigned)(unsigned long long)(void*)tile;  // LDS byte offset

  // ---------------- Phase A + B: DMA tile(s) in, reduce per-dim min/max ----
  float mn = FLT_BIG, mx = -FLT_BIG;
  for (int rbase = 0; rbase < cnt; rbase += CHUNK) {
    const int rows = min(CHUNK, cnt - rbase);
    const int nvec = rows * VEC_PER_ROW;
    for (int c = tid; c < nvec; c += TPB) {
      async_copy_b128(lds0 + ((unsigned)c << 4),
                      (unsigned)(rbase * VEC_PER_ROW + c) << 4, g_in);
    }
    wait_async0();
    __syncthreads();
    if (tid < D) {
      for (int r = 0; r < rows; ++r) {
        const float v = tile[r * D + tid];   // lane-consecutive dims: no bank conflicts
        mn = fminf(mn, v);
        mx = fmaxf(mx, v);
      }
    }
    __syncthreads();  // tile must be fully read before any re-fill
  }

  if (tid < D) {
    const float mid = (mx + mn) * 0.5f;
    float ldv = (mx - mn) * 0.5f;
    ldv = fmaxf(ldv, EPS);
    s_mid[tid] = mid;
    s_inv[tid] = 1.0f / ldv;
  }
  __syncthreads();

  const float4* __restrict__ midv = (const float4*)s_mid;
  const float4* __restrict__ invv = (const float4*)s_inv;
  float4* __restrict__ outv = (float4*)g_out;

  // ---------------- Phase C: normalize ------------------------------------
  if (cnt <= CHUNK) {
    // Fast path (actual input): whole graph is still resident in LDS.
    const int nvec = cnt * VEC_PER_ROW;
    const float4* __restrict__ tv = (const float4*)tile;
    int cm = tid % VEC_PER_ROW;                 // chunk index within row
    for (int c = tid; c < nvec; c += TPB) {
      const float4 v = tv[c];
      const float4 m = midv[cm];
      const float4 s = invv[cm];
      float4 r;
      r.x = (v.x - m.x) * s.x;
      r.y = (v.y - m.y) * s.y;
      r.z = (v.z - m.z) * s.z;
      r.w = (v.w - m.w) * s.w;
      outv[c] = r;                              // global_store_b128
      cm += (TPB % VEC_PER_ROW);                // incremental mod-75
      if (cm >= VEC_PER_ROW) cm -= VEC_PER_ROW;
    }
  } else {
    // General path: re-stream tiles through LDS.
    for (int rbase = 0; rbase < cnt; rbase += CHUNK) {
      const int rows  = min(CHUNK, cnt - rbase);
      const int nvec  = rows * VEC_PER_ROW;
      const int cbase = rbase * VEC_PER_ROW;    // multiple of VEC_PER_ROW
      for (int c = tid; c < nvec; c += TPB) {
        async_copy_b128(lds0 + ((unsigned)c << 4),
                        (unsigned)(cbase + c) << 4, g_in);
      }
      wait_async0();
      __syncthreads();
      const float4* __restrict__ tv = (const float4*)tile;
      int cm = tid % VEC_PER_ROW;
      for (int c = tid; c < nvec; c += TPB) {
        const float4 v = tv[c];
        const float4 m = midv[cm];
        const float4 s = invv[cm];
        float4 r;
        r.x = (v.x - m.x) * s.x;
        r.y = (v.y - m.y) * s.y;
        r.z = (v.z - m.z) * s.z;
        r.w = (v.w - m.w) * s.w;
        outv[cbase + c] = r;
        cm += (TPB % VEC_PER_ROW);
        if (cm >= VEC_PER_ROW) cm -= VEC_PER_ROW;
      }
      __syncthreads();
    }
  }
}

// ---------------------------------------------------------------------------
extern "C" void kernel_launch(void* const* d_in, const int* in_sizes, int n_in,
                              void* d_out, int out_size, void* d_ws, size_t ws_size,
                              hipStream_t stream) {
  const float* x      = (const float*)d_in[0];
  const int*   counts = (const int*)d_in[1];   // batch_list (JAX default: int32)
  const int    B      = in_sizes[1];
  float*       out    = (float*)d_out;
  int*         offs   = (int*)d_ws;            // B ints of scratch

  exclusive_scan_kernel<<<dim3(1), dim3(SCAN_MAX), 0, stream>>>(counts, offs, B);
  seg_minmax_norm_kernel<<<dim3(B), dim3(TPB), 0, stream>>>(x, counts, offs, out, B);
}